// KBpA_1073741824066
// MI455X (gfx1250) — compile-verified
//
#include <hip/hip_runtime.h>

// ---------------------------------------------------------------------------
// out = exp(-pairwise_sqdist(pixels - K@betas, centers) / (2*4)) @ alphas
// N_PIXELS = 65536, N_CENTERS = 4096, SDP2 = 4
// ---------------------------------------------------------------------------

#define N_PIXELS  65536
#define N_CENTERS 4096
#define N_CTILES  (N_CENTERS / 16)   // 256
#define N_PTILES  (N_PIXELS / 16)    // 4096
#define SPLIT     4                  // center-dimension split for parallelism
#define TILES_PER_SPLIT (N_CTILES / SPLIT)

// exp(-d / (2*SDP2)) == exp2(d * KEXP),  KEXP = -log2(e)/8
#define KEXP (-0.18033688011112042592f)

typedef float v2f __attribute__((ext_vector_type(2)));
typedef float v8f __attribute__((ext_vector_type(8)));

// ---------------------------------------------------------------------------
// Kernel 0: build WMMA B-operand staging, lane-ordered per 16-center tile.
// B is 4x16 (KxN). Per ISA: VGPR0 holds K=0 (lanes 0-15) / K=2 (lanes 16-31),
// VGPR1 holds K=1 / K=3.  Column N = lane%16.
//   K0 = -2*cx, K1 = -2*cy, K2 = 1, K3 = cx^2+cy^2
// Bstage[tile*32 + lane] = float4(b_vgpr0, b_vgpr1, alpha[N], 0)
// ---------------------------------------------------------------------------
__global__ void prep_centers_kernel(const float* __restrict__ centers,
                                    const float* __restrict__ alphas,
                                    float4* __restrict__ Bstage) {
    int gid  = blockIdx.x * blockDim.x + threadIdx.x;   // 0 .. 256*32-1
    int tile = gid >> 5;
    int lane = gid & 31;
    int c    = tile * 16 + (lane & 15);
    float cx = centers[2 * c + 0];
    float cy = centers[2 * c + 1];
    float b0, b1;
    if (lane < 16) { b0 = -2.0f * cx; b1 = -2.0f * cy; }
    else           { b0 = 1.0f;       b1 = cx * cx + cy * cy; }
    Bstage[gid] = make_float4(b0, b1, alphas[c], 0.0f);
}

// ---------------------------------------------------------------------------
// Kernel 1: deformation = K_dense @ betas  (HBM-bound: reads the 1 GiB input).
// One wave per pixel; lane-strided coalesced reads of the 4096-wide row.
// Writes Pstage[p] = (dx, dy, dx^2+dy^2, 1)  -- the WMMA A-operand payload.
// ---------------------------------------------------------------------------
__global__ void deform_kernel(const float* __restrict__ K,
                              const float* __restrict__ betas,
                              const float* __restrict__ pixels,
                              float4* __restrict__ Pstage) {
    const int lane = threadIdx.x & 31;
    const int p    = (blockIdx.x * blockDim.x + threadIdx.x) >> 5;  // pixel id
    const float* __restrict__ row = K + (size_t)p * N_CENTERS;
    const float2* __restrict__ b2 = (const float2*)betas;

    float ax = 0.0f, ay = 0.0f;
#pragma unroll 4
    for (int c = lane; c < N_CENTERS; c += 32) {
        float  k = row[c];
        float2 b = b2[c];
        ax = fmaf(k, b.x, ax);
        ay = fmaf(k, b.y, ay);
    }
#pragma unroll
    for (int off = 16; off >= 1; off >>= 1) {
        ax += __shfl_xor(ax, off, 32);
        ay += __shfl_xor(ay, off, 32);
    }
    if (lane == 0) {
        float dx = pixels[2 * p + 0] - ax;
        float dy = pixels[2 * p + 1] - ay;
        Pstage[p] = make_float4(dx, dy, dx * dx + dy * dy, 1.0f);
    }
}

// ---------------------------------------------------------------------------
// Kernel 2: fused RBF + alpha-reduction via V_WMMA_F32_16X16X4_F32.
// One wave handles a 16-pixel tile for one center-split (64 center tiles).
// A (16x4) per ISA layout: lanes 0-15: VGPR0=K0=dx, VGPR1=K1=dy (M=lane);
//                          lanes 16-31: VGPR0=K2=|dp|^2, VGPR1=K3=1 (M=lane-16).
// One WMMA = full 16x16 tile of squared distances; then 8x(mul,exp2,fma).
// C/D layout: element (M = i + 8*(lane/16), N = lane%16) in acc[i] -> the
// alpha for a lane's column is s.z (constant over i). Butterfly-reduce over
// N (xor 1,2,4,8 stays inside each 16-lane half); lanes 0/16 own rows 0-7/8-15.
// ---------------------------------------------------------------------------
__global__ void rbf_kernel(const float4* __restrict__ Bstage,
                           const float4* __restrict__ Pstage,
                           float* __restrict__ partial) {
    const int lane    = threadIdx.x & 31;
    const int wave    = (blockIdx.x * blockDim.x + threadIdx.x) >> 5;
    const int pixTile = wave >> 2;          // wave / SPLIT
    const int split   = wave & (SPLIT - 1);

    float4 P = Pstage[pixTile * 16 + (lane & 15)];
    v2f a;
    a.x = (lane < 16) ? P.x : P.z;
    a.y = (lane < 16) ? P.y : P.w;

    v8f acc = {};
    const int tBeg = split * TILES_PER_SPLIT;
    const int tEnd = tBeg + TILES_PER_SPLIT;
    for (int t = tBeg; t < tEnd; ++t) {
        float4 s = Bstage[t * 32 + lane];
        v2f b;
        b.x = s.x;
        b.y = s.y;
        v8f c = {};
        c = __builtin_amdgcn_wmma_f32_16x16x4_f32(
                /*neg_a=*/false, a, /*neg_b=*/false, b,
                /*c_mod=*/(short)0, c, /*reuse_a=*/false, /*reuse_b=*/false);
#pragma unroll
        for (int i = 0; i < 8; ++i) {
            float e = __builtin_amdgcn_exp2f(c[i] * KEXP);  // v_exp_f32
            acc[i]  = fmaf(s.z, e, acc[i]);
        }
    }

#pragma unroll
    for (int i = 0; i < 8; ++i) {
#pragma unroll
        for (int off = 1; off < 16; off <<= 1)
            acc[i] += __shfl_xor(acc[i], off, 32);
    }

    if ((lane & 15) == 0) {
        int rowBase = pixTile * 16 + ((lane >> 4) << 3);   // lane 0 -> rows 0-7, lane 16 -> rows 8-15
        float* dst = partial + (size_t)split * N_PIXELS + rowBase;
#pragma unroll
        for (int i = 0; i < 8; ++i) dst[i] = acc[i];
    }
}

// ---------------------------------------------------------------------------
// Kernel 3: deterministic combine of the SPLIT partials.
// ---------------------------------------------------------------------------
__global__ void reduce_kernel(const float* __restrict__ partial,
                              float* __restrict__ out) {
    int p = blockIdx.x * blockDim.x + threadIdx.x;
    float s = 0.0f;
#pragma unroll
    for (int k = 0; k < SPLIT; ++k) s += partial[(size_t)k * N_PIXELS + p];
    out[p] = s;
}

// ---------------------------------------------------------------------------
extern "C" void kernel_launch(void* const* d_in, const int* in_sizes, int n_in,
                              void* d_out, int out_size, void* d_ws, size_t ws_size,
                              hipStream_t stream) {
    const float* K_dense = (const float*)d_in[0];   // [65536, 4096]
    const float* betas   = (const float*)d_in[1];   // [4096, 2]
    const float* alphas  = (const float*)d_in[2];   // [4096, 1]
    const float* pixels  = (const float*)d_in[3];   // [65536, 2]
    const float* centers = (const float*)d_in[4];   // [4096, 2]
    float*       out     = (float*)d_out;           // [65536, 1]

    // Workspace layout (floats): Bstage | Pstage | partial
    float*  ws      = (float*)d_ws;
    float4* Bstage  = (float4*)(ws);                          // 256*32*4  = 32768 f (128 KB)
    float4* Pstage  = (float4*)(ws + N_CTILES * 32 * 4);      // 65536*4   = 262144 f (1 MB)
    float*  partial = ws + N_CTILES * 32 * 4 + N_PIXELS * 4;  // 4*65536   = 262144 f (1 MB)

    // 0) pack B-operand staging: 256 tiles * 32 lanes = 8192 threads
    prep_centers_kernel<<<(N_CTILES * 32) / 256, 256, 0, stream>>>(centers, alphas, Bstage);

    // 1) deformation (1 GiB read): one wave per pixel -> 65536 waves
    deform_kernel<<<(N_PIXELS * 32) / 256, 256, 0, stream>>>(K_dense, betas, pixels, Pstage);

    // 2) fused WMMA distance + exp + alpha partial sums: 4096*SPLIT waves
    rbf_kernel<<<(N_PTILES * SPLIT * 32) / 256, 256, 0, stream>>>(Bstage, Pstage, partial);

    // 3) combine split partials
    reduce_kernel<<<N_PIXELS / 256, 256, 0, stream>>>(partial, out);
}